// TCR_52536039964687
// MI455X (gfx1250) — compile-verified
//
#include <hip/hip_runtime.h>

// Problem constants (img: 64 x 3 x 256 x 256 fp32)
#define IMG_W  256
#define IMG_H  256
#define IMG_C  3
#define NBATCH 64
#define TILE   32   // output tile per workgroup (32x32)
#define SMAX   64   // staged window stride cap (bw+1 <= 64; worst need ~58)

typedef unsigned int v4u __attribute__((ext_vector_type(4)));
typedef int          v8i __attribute__((ext_vector_type(8)));
typedef int          v4i __attribute__((ext_vector_type(4)));

// Minimax-grade polynomials, valid for |x| <= ~0.8 rad (our args are <= 0.7).
// Avoids libm's Payne-Hanek slow path (~1K dead SALU instructions).
__device__ __forceinline__ float psin(float x) {
    float x2 = x * x;
    return x * (1.f + x2 * (-1.6666667e-1f +
                x2 * (8.3333333e-3f + x2 * (-1.9841270e-4f))));
}
__device__ __forceinline__ float pcos(float x) {
    float x2 = x * x;
    return 1.f + x2 * (-0.5f + x2 * (4.1666667e-2f +
                x2 * (-1.3888889e-3f + x2 * 2.4801587e-5f)));
}

// One workgroup (256 threads = 8 waves) produces a 32x32 output tile for all
// 3 channels of one batch image. The input window (plus a +1 zero-filled rim,
// courtesy of TDM OOB semantics) is staged into LDS by the Tensor Data Mover;
// bilinear taps then pair into ds_load_2addr, and each thread writes one
// float4 per channel.
__global__ __launch_bounds__(256) void warp_affine_tdm(
    const float* __restrict__ img, const float* __restrict__ rnd,
    float* __restrict__ out)
{
    __shared__ float smem[IMG_C * SMAX * SMAX];   // 48 KB

    const int b   = blockIdx.z;
    const int tx0 = blockIdx.x * TILE;
    const int ty0 = blockIdx.y * TILE;

    // ---- per-batch affine transform + inverse (block-uniform -> SALU) ----
    const float ANG  = 0.3490658503988659f;  // deg2rad(20)
    const float MAXT = 6.0f;
    const float r  = rnd[b];
    const float tx = 2.f * MAXT * r - MAXT;
    const float ty = 2.f * MAXT * r - MAXT;
    const float ro = 2.f * ANG  * r - ANG;
    const float hx = 2.f * ANG  * r - ANG;
    const float hy = 2.f * ANG  * r - ANG;
    const float aa = hx - ro;                 // == 0 (same r), kept general
    const float bb = hy + ro;                 // |bb| <= 0.7 rad
    const float chx = pcos(hx), chy = pcos(hy);
    const float ca = pcos(aa), sa = psin(aa);
    const float cb = pcos(bb), sb = psin(bb);
    const float Wc = (float)IMG_H;   // img.shape[2]
    const float Hc = (float)IMG_W;   // img.shape[3]
    const float T11 = ca / chx;
    const float T12 = sa / chx;
    const float T13 = (Wc*chx - Wc*ca + 2.f*tx*ca - Hc*sa + 2.f*ty*sa) / (2.f*chx);
    const float T21 = sb / chy;
    const float T22 = cb / chy;
    const float T23 = (Hc*chy - Wc*cb + 2.f*ty*cb - Wc*sb + 2.f*tx*sb) / (2.f*chy);
    const float det = T11*T22 - T12*T21;
    const float ia  =  T22 / det;
    const float ib  = -T12 / det;
    const float idc = -T21 / det;
    const float ie  =  T11 / det;
    const float ic  = -(ia  * T13 + ib * T23);
    const float iff = -(idc * T13 + ie * T23);

    // ---- input bbox of this output tile (affine => corners suffice) ----
    const float x0f = (float)tx0, x1f = (float)(tx0 + TILE - 1);
    const float y0f = (float)ty0, y1f = (float)(ty0 + TILE - 1);
    float sxc0 = ia*x0f + ib*y0f + ic,   sxc1 = ia*x1f + ib*y0f + ic;
    float sxc2 = ia*x0f + ib*y1f + ic,   sxc3 = ia*x1f + ib*y1f + ic;
    float syc0 = idc*x0f + ie*y0f + iff, syc1 = idc*x1f + ie*y0f + iff;
    float syc2 = idc*x0f + ie*y1f + iff, syc3 = idc*x1f + ie*y1f + iff;
    float minsx = fminf(fminf(sxc0, sxc1), fminf(sxc2, sxc3));
    float maxsx = fmaxf(fmaxf(sxc0, sxc1), fmaxf(sxc2, sxc3));
    float minsy = fminf(fminf(syc0, syc1), fminf(syc2, syc3));
    float maxsy = fmaxf(fmaxf(syc0, syc1), fmaxf(syc2, syc3));

    int bx = (int)floorf(minsx) - 1;
    int by = (int)floorf(minsy) - 1;
    int ex = (int)floorf(maxsx) + 3;   // +1 for x0+1 tap, +margin
    int ey = (int)floorf(maxsy) + 3;
    bx = min(max(bx, 0), IMG_W - 1);
    by = min(max(by, 0), IMG_H - 1);
    ex = min(ex, IMG_W);
    ey = min(ey, IMG_H);
    const int bw  = min(max(ex - bx, 1), SMAX - 1);  // valid cols [0, bw-1]
    const int bh  = min(max(ey - by, 1), SMAX - 1);  // valid rows [0, bh-1]
    const int bws = bw + 1;                          // staged row stride
    const int bhs = bh + 1;                          // staged rows
    const int chs = bws * bhs;                       // floats per channel

    // ---- TDM: stage (bw+1) x (bh+1) x 3 into LDS; OOB rim zero-fills ----
    if (threadIdx.x == 0) {
        unsigned long long gaddr = (unsigned long long)(uintptr_t)
            (img + (size_t)b * IMG_C * IMG_H * IMG_W + (size_t)by * IMG_W + bx);
        unsigned lds_off = (unsigned)(size_t)(void*)smem;   // LDS byte offset

        // D# group 0: count=1 | lds_addr | global_addr | type=2
        v4u g0;
        g0[0] = 1u;
        g0[1] = lds_off;
        g0[2] = (unsigned)(gaddr & 0xFFFFFFFFull);
        g0[3] = (unsigned)((gaddr >> 32) & 0x01FFFFFFull) | 0x80000000u;

        // D# group 1: data_size=4B; tensor dims bound the valid extent so the
        // +1 rim beyond the image returns zeros; tile = bws x bhs x 3;
        // strides 256 / 65536 elements.
        const unsigned td0 = (unsigned)(IMG_W - bx);
        const unsigned td1 = (unsigned)(IMG_H - by);
        const unsigned long long s0 = (unsigned long long)IMG_W;          // dim0 stride
        const unsigned long long s1 = (unsigned long long)(IMG_H*IMG_W);  // dim1 stride
        v8i g1;
        g1[0] = (int)(2u << 16);                                   // data_size=2 (4B)
        g1[1] = (int)((td0 & 0xFFFFu) << 16);                      // tensor_dim0 lo
        g1[2] = (int)((td0 >> 16) | ((td1 & 0xFFFFu) << 16));      // td0 hi | td1 lo
        g1[3] = (int)((td1 >> 16) | ((unsigned)bws << 16));        // td1 hi | tile_dim0
        g1[4] = (int)((unsigned)bhs | (3u << 16));                 // tile_dim1 | tile_dim2
        g1[5] = (int)(s0 & 0xFFFFFFFFull);                         // dim0 stride lo
        g1[6] = (int)(((s0 >> 32) & 0xFFFFull) | ((s1 & 0xFFFFull) << 16));
        g1[7] = (int)((s1 >> 16) & 0xFFFFFFFFull);                 // dim1 stride hi

        v4i g2 = {IMG_C, 0, 0, 0};   // tensor_dim2 = 3; tile_dim3 unused
        v4i g3 = {0, 0, 0, 0};

#if defined(__clang_major__) && (__clang_major__ >= 23)
        v8i g4 = {0, 0, 0, 0, 0, 0, 0, 0};
        __builtin_amdgcn_tensor_load_to_lds(g0, g1, g2, g3, g4, 0);
#else
        __builtin_amdgcn_tensor_load_to_lds(g0, g1, g2, g3, 0);
#endif
        __builtin_amdgcn_s_wait_tensorcnt(0);
    }
    __syncthreads();

    // ---- bilinear sampling from LDS ----
    // Thread t handles row (t>>3), columns 4*(t&7) .. 4*(t&7)+3 of the tile.
    // Taps (i00,i00+1) and (i10,i10+1) are adjacent -> ds_load_2addr pairs.
    const int row = threadIdx.x >> 3;
    const int xq  = (threadIdx.x & 7) * 4;
    const int yy  = ty0 + row;
    const int xxb = tx0 + xq;

    const float sx0 = ia  * (float)xxb + ib * (float)yy + ic;
    const float sy0 = idc * (float)xxb + ie * (float)yy + iff;

    float res[IMG_C][4];

#pragma unroll
    for (int k = 0; k < 4; ++k) {
        const float sx = sx0 + ia  * (float)k;
        const float sy = sy0 + idc * (float)k;
        const float xf = floorf(sx), yf = floorf(sy);
        const int   x0 = (int)xf,    y0 = (int)yf;
        const float wx = sx - xf,    wy = sy - yf;
        const float w00 = (1.f - wx) * (1.f - wy);
        const float w01 = wx * (1.f - wy);
        const float w10 = (1.f - wx) * wy;
        const float w11 = wx * wy;

        const bool vx0 = (x0 >= 0)     && (x0 < IMG_W);
        const bool vx1 = (x0 + 1 >= 0) && (x0 + 1 < IMG_W);
        const bool vy0 = (y0 >= 0)     && (y0 < IMG_H);
        const bool vy1 = (y0 + 1 >= 0) && (y0 + 1 < IMG_H);
        const float c00 = (vx0 && vy0) ? w00 : 0.f;
        const float c01 = (vx1 && vy0) ? w01 : 0.f;
        const float c10 = (vx0 && vy1) ? w10 : 0.f;
        const float c11 = (vx1 && vy1) ? w11 : 0.f;

        // Base tap clamped into the valid window; the +1 / +stride taps stay
        // inside the staged (bws x bhs) region, which is fully TDM-written
        // (zero-filled where it crosses the image edge).
        const int lx0 = min(max(x0 - bx, 0), bw - 1);
        const int ly0 = min(max(y0 - by, 0), bh - 1);
        const int i00 = ly0 * bws + lx0;
        const int i10 = i00 + bws;

#pragma unroll
        for (int c = 0; c < IMG_C; ++c) {
            const float* sc = smem + c * chs;
            res[c][k] = sc[i00] * c00 + sc[i00 + 1] * c01 +
                        sc[i10] * c10 + sc[i10 + 1] * c11;
        }
    }

#pragma unroll
    for (int c = 0; c < IMG_C; ++c) {
        float4 v = make_float4(res[c][0], res[c][1], res[c][2], res[c][3]);
        float4* dst = (float4*)(out +
            (((size_t)b * IMG_C + c) * IMG_H + yy) * IMG_W + xxb);
        *dst = v;   // global_store_b128, 16B-aligned (xxb % 4 == 0)
    }
}

extern "C" void kernel_launch(void* const* d_in, const int* in_sizes, int n_in,
                              void* d_out, int out_size, void* d_ws, size_t ws_size,
                              hipStream_t stream) {
    (void)in_sizes; (void)n_in; (void)d_ws; (void)ws_size; (void)out_size;
    const float* img = (const float*)d_in[0];
    const float* rnd = (const float*)d_in[1];
    float* out = (float*)d_out;
    dim3 grid(IMG_W / TILE, IMG_H / TILE, NBATCH);   // 8 x 8 x 64
    dim3 block(256);
    warp_affine_tdm<<<grid, block, 0, stream>>>(img, rnd, out);
}